// GraphCON_GAT_65292092834214
// MI455X (gfx1250) — compile-verified
//
#include <hip/hip_runtime.h>
#include <hip/hip_bf16.h>
#include <math.h>

#define N_NODES 50000
#define N_EDGES 800000
#define ET (N_EDGES + N_NODES)   /* edges + self loops */
#define N_GRAPHS 64
#define NFEAT 128
#define NHID 64
#define NHEADS 4
#define HW (NHEADS * NHID)       /* 256 */
#define NLAYERS 3
#define NEG_SLOPE 0.2f
#define DT_C 1.0f
#define ALPHA_C 1.0f
#define GAMMA_C 1.0f

typedef __attribute__((ext_vector_type(2))) float v2f;
typedef __attribute__((ext_vector_type(8))) float v8f;

// ---------------------------------------------------------------- utilities
__device__ __forceinline__ void atomicAddF(float* addr, float v) {
  __hip_atomic_fetch_add(addr, v, __ATOMIC_RELAXED, __HIP_MEMORY_SCOPE_AGENT);
}

__device__ __forceinline__ void atomicMaxF(float* addr, float v) {
  // IEEE-754 ordering trick: signed-int max for v>=0, unsigned-int min for v<0
  if (v >= 0.0f) {
    __hip_atomic_fetch_max((int*)addr, __float_as_int(v),
                           __ATOMIC_RELAXED, __HIP_MEMORY_SCOPE_AGENT);
  } else {
    __hip_atomic_fetch_min((unsigned int*)addr, __float_as_uint(v),
                           __ATOMIC_RELAXED, __HIP_MEMORY_SCOPE_AGENT);
  }
}

__global__ void fill_f32(float* p, float v, int n) {
  int t = blockIdx.x * blockDim.x + threadIdx.x;
  if (t < n) p[t] = v;
}

__global__ void copy_f32(const float* __restrict__ s, float* __restrict__ d, int n) {
  int t = blockIdx.x * blockDim.x + threadIdx.x;
  if (t < n) d[t] = s[t];
}

// ----------------------------------------------- input concat: [x|pos] -> inp
__global__ void concat_kernel(const float* __restrict__ x, const float* __restrict__ pos,
                              float* __restrict__ inp) {
  int t = blockIdx.x * blockDim.x + threadIdx.x;   // n*128 + c
  if (t >= N_NODES * NFEAT) return;
  int n = t >> 7, c = t & 127;
  inp[t] = (c < 125) ? x[n * 125 + c] : pos[n * 3 + (c - 125)];
}

// ------------------------------------------------------- fp32 WMMA GEMM
// Out[M,NC] = A[M,K] @ W[K,NC] (+ bias). One wave computes one 16x16 tile
// using V_WMMA_F32_16X16X4_F32.  M, NC multiples of 16; K multiple of 4.
__global__ void gemm_wmma_f32(const float* __restrict__ A, const float* __restrict__ W,
                              const float* __restrict__ bias, float* __restrict__ Out,
                              int M, int K, int NC) {
  int wave = (int)((blockIdx.x * blockDim.x + threadIdx.x) >> 5);
  int lane = threadIdx.x & 31;
  int tiles_n = NC >> 4;
  int tiles_total = (M >> 4) * tiles_n;
  if (wave >= tiles_total) return;          // whole-wave exit: EXEC stays all-1s
  int tm = wave / tiles_n, tn = wave % tiles_n;
  int m0 = tm << 4, n0 = tn << 4;
  int half = lane >> 4;                     // 0: K pair {0,1}, 1: K pair {2,3}
  int lm = lane & 15;

  const float* arow = A + (size_t)(m0 + lm) * K;
  v8f acc = {};
  for (int k = 0; k < K; k += 4) {
    int kb = k + (half << 1);               // even -> 8B aligned
    v2f a = *(const v2f*)(arow + kb);       // A[m][kb], A[m][kb+1]
    v2f b;
    b.x = W[(size_t)kb * NC + n0 + lm];     // B[kb][n]
    b.y = W[(size_t)(kb + 1) * NC + n0 + lm];
    acc = __builtin_amdgcn_wmma_f32_16x16x4_f32(
        /*neg_a=*/false, a, /*neg_b=*/false, b,
        /*c_mod=*/(short)0, acc, /*reuse_a=*/false, /*reuse_b=*/false);
  }
  float bv = bias ? bias[n0 + lm] : 0.0f;
#pragma unroll
  for (int r = 0; r < 8; ++r)               // C/D: VGPR r -> row r + 8*half
    Out[(size_t)(m0 + r + (half << 3)) * NC + n0 + lm] = acc[r] + bv;
}

// ---------------------------------- per-node attention projections a_src/a_dst
__global__ void attn_proj_kernel(const float* __restrict__ h,
                                 const float* __restrict__ att_src,
                                 const float* __restrict__ att_dst,
                                 float* __restrict__ a_src, float* __restrict__ a_dst) {
  int t = blockIdx.x * blockDim.x + threadIdx.x;   // n*4 + head
  if (t >= N_NODES * NHEADS) return;
  int n = t >> 2, hd = t & 3;
  const float* hp = h + (size_t)n * HW + hd * NHID;
  const float* ws = att_src + hd * NHID;
  const float* wd = att_dst + hd * NHID;
  float ss = 0.f, sd = 0.f;
#pragma unroll 8
  for (int c = 0; c < NHID; ++c) { ss += hp[c] * ws[c]; sd += hp[c] * wd[c]; }
  a_src[t] = ss;
  a_dst[t] = sd;
}

// ------------------------------------- edge pass 1: leaky-relu logit + seg-max
__global__ void edge_logits_max(const int* __restrict__ eidx,
                                const float* __restrict__ a_src,
                                const float* __restrict__ a_dst,
                                float* __restrict__ ebuf, float* __restrict__ m) {
  int t = blockIdx.x * blockDim.x + threadIdx.x;   // e*4 + head
  if (t >= ET * NHEADS) return;
  int e = t >> 2, hd = t & 3;
  int s, d;
  if (e < N_EDGES) { s = eidx[e]; d = eidx[N_EDGES + e]; }
  else             { s = d = e - N_EDGES; }        // self loop
  float v = a_src[s * NHEADS + hd] + a_dst[d * NHEADS + hd];
  v = (v > 0.f) ? v : NEG_SLOPE * v;
  ebuf[t] = v;
  atomicMaxF(&m[d * NHEADS + hd], v);
}

// ------------------------------------- edge pass 2: exp(e - m[dst]) + seg-sum
__global__ void edge_exp_sum(const int* __restrict__ eidx,
                             const float* __restrict__ m,
                             float* __restrict__ ebuf, float* __restrict__ z) {
  int t = blockIdx.x * blockDim.x + threadIdx.x;
  if (t >= ET * NHEADS) return;
  int e = t >> 2, hd = t & 3;
  int d = (e < N_EDGES) ? eidx[N_EDGES + e] : (e - N_EDGES);
  float p = expf(ebuf[t] - m[d * NHEADS + hd]);
  ebuf[t] = p;
  atomicAddF(&z[d * NHEADS + hd], p);
}

// --------------------- edge pass 3: att = p/z[dst]; conv[dst] += h[src]*att
// One wave per edge; lane j covers columns j, j+32, ..., j+224 (head = t>>1).
__global__ void edge_scatter(const int* __restrict__ eidx,
                             const float* __restrict__ h,
                             const float* __restrict__ ebuf,
                             const float* __restrict__ z,
                             float* __restrict__ conv) {
  int wave = (int)((blockIdx.x * blockDim.x + threadIdx.x) >> 5);
  int lane = threadIdx.x & 31;
  if (wave >= ET) return;
  int e = wave, s, d;
  if (e < N_EDGES) { s = eidx[e]; d = eidx[N_EDGES + e]; }
  else             { s = d = e - N_EDGES; }
  const float* hrow = h + (size_t)s * HW;
  __builtin_prefetch(hrow + lane * 8, 0, 1);       // global_prefetch_b8
  float att[NHEADS];
#pragma unroll
  for (int hd = 0; hd < NHEADS; ++hd)
    att[hd] = ebuf[e * NHEADS + hd] / z[d * NHEADS + hd];
  float* crow = conv + (size_t)d * HW;
#pragma unroll
  for (int t = 0; t < 8; ++t) {
    int col = lane + (t << 5);                     // head = col>>6 == t>>1
    atomicAddF(&crow[col], hrow[col] * att[t >> 1]);
  }
}

// ------------- combine: mixed = elu(conv + conv_b + res) head-mix; ODE update
__global__ void combine_update(const float* __restrict__ conv,
                               const float* __restrict__ conv_b,
                               const float* __restrict__ res,
                               float* __restrict__ Y, float* __restrict__ X) {
  int t = blockIdx.x * blockDim.x + threadIdx.x;   // n*64 + i
  if (t >= N_NODES * NHID) return;
  int n = t >> 6, i = t & 63;
  const float* c = conv + (size_t)n * HW + i * 4;
  const float* r = res  + (size_t)n * HW + i * 4;
  const float* cb = conv_b + i * 4;
  float s = 0.f;
#pragma unroll
  for (int j = 0; j < 4; ++j) {
    float v = c[j] + cb[j] + r[j];
    s += (v > 0.f) ? v : (expf(v) - 1.0f);         // ELU
  }
  float mixed = 0.25f * s;
  float y = Y[t], x = X[t];
  float yn = y + DT_C * (mixed - ALPHA_C * y - GAMMA_C * x);
  Y[t] = yn;
  X[t] = x + DT_C * yn;
}

// ----------------------------------------- decoder + per-graph pooled sum
__global__ void decode_pool(const float* __restrict__ X,
                            const float* __restrict__ dec_w,
                            const float* __restrict__ dec_b,
                            const int* __restrict__ batch,
                            float* __restrict__ out) {
  int n = blockIdx.x * blockDim.x + threadIdx.x;
  if (n >= N_NODES) return;
  const float* xr = X + (size_t)n * NHID;
  float acc = dec_b[0];
#pragma unroll 8
  for (int c = 0; c < NHID; ++c) acc += xr[c] * dec_w[c];
  atomicAddF(&out[batch[n]], acc);
}

// ---------------------------------------------------------------- launcher
extern "C" void kernel_launch(void* const* d_in, const int* in_sizes, int n_in,
                              void* d_out, int out_size, void* d_ws, size_t ws_size,
                              hipStream_t stream) {
  const float* x       = (const float*)d_in[0];
  const float* pos     = (const float*)d_in[1];
  const int*   eidx    = (const int*)d_in[2];
  const int*   batch   = (const int*)d_in[3];
  const float* enc_w   = (const float*)d_in[4];
  const float* enc_b   = (const float*)d_in[5];
  const float* res_w   = (const float*)d_in[6];
  const float* res_b   = (const float*)d_in[7];
  const float* lin_w   = (const float*)d_in[8];
  const float* att_src = (const float*)d_in[9];
  const float* att_dst = (const float*)d_in[10];
  const float* conv_b  = (const float*)d_in[11];
  const float* dec_w   = (const float*)d_in[12];
  const float* dec_b   = (const float*)d_in[13];
  float* out = (float*)d_out;

  // -------- workspace carve
  size_t off = 0;
  auto carve = [&](size_t bytes) -> float* {
    float* p = (float*)((char*)d_ws + off);
    off += (bytes + 255) & ~(size_t)255;
    return p;
  };
  float* inp   = carve((size_t)N_NODES * NFEAT * 4);
  float* Y     = carve((size_t)N_NODES * NHID * 4);
  float* Xb    = carve((size_t)N_NODES * NHID * 4);
  float* hbuf  = carve((size_t)N_NODES * HW * 4);
  float* resb  = carve((size_t)N_NODES * HW * 4);
  float* convb = carve((size_t)N_NODES * HW * 4);
  float* a_src = carve((size_t)N_NODES * NHEADS * 4);
  float* a_dst = carve((size_t)N_NODES * NHEADS * 4);
  float* mbuf  = carve((size_t)N_NODES * NHEADS * 4);
  float* zbuf  = carve((size_t)N_NODES * NHEADS * 4);
  float* ebuf  = carve((size_t)ET * NHEADS * 4);
  (void)ws_size; (void)n_in; (void)in_sizes; (void)out_size;

  const int B = 256;
  auto nb = [](long long n, int b) { return (int)((n + b - 1) / b); };

  // -------- encoder
  concat_kernel<<<nb((long long)N_NODES * NFEAT, B), B, 0, stream>>>(x, pos, inp);
  {
    int tiles = (N_NODES / 16) * (NHID / 16);
    gemm_wmma_f32<<<nb((long long)tiles * 32, B), B, 0, stream>>>(
        inp, enc_w, enc_b, Y, N_NODES, NFEAT, NHID);
  }
  copy_f32<<<nb((long long)N_NODES * NHID, B), B, 0, stream>>>(Y, Xb, N_NODES * NHID);

  // -------- 3 GraphCON-GAT layers
  for (int layer = 0; layer < NLAYERS; ++layer) {
    int tiles = (N_NODES / 16) * (HW / 16);
    gemm_wmma_f32<<<nb((long long)tiles * 32, B), B, 0, stream>>>(
        Xb, lin_w, nullptr, hbuf, N_NODES, NHID, HW);
    gemm_wmma_f32<<<nb((long long)tiles * 32, B), B, 0, stream>>>(
        Xb, res_w, res_b, resb, N_NODES, NHID, HW);
    attn_proj_kernel<<<nb((long long)N_NODES * NHEADS, B), B, 0, stream>>>(
        hbuf, att_src, att_dst, a_src, a_dst);

    fill_f32<<<nb((long long)N_NODES * NHEADS, B), B, 0, stream>>>(
        mbuf, -3.0e38f, N_NODES * NHEADS);
    fill_f32<<<nb((long long)N_NODES * NHEADS, B), B, 0, stream>>>(
        zbuf, 0.0f, N_NODES * NHEADS);
    fill_f32<<<nb((long long)N_NODES * HW, B), B, 0, stream>>>(
        convb, 0.0f, N_NODES * HW);

    edge_logits_max<<<nb((long long)ET * NHEADS, B), B, 0, stream>>>(
        eidx, a_src, a_dst, ebuf, mbuf);
    edge_exp_sum<<<nb((long long)ET * NHEADS, B), B, 0, stream>>>(
        eidx, mbuf, ebuf, zbuf);
    edge_scatter<<<nb((long long)ET * 32, B), B, 0, stream>>>(
        eidx, hbuf, ebuf, zbuf, convb);

    combine_update<<<nb((long long)N_NODES * NHID, B), B, 0, stream>>>(
        convb, conv_b, resb, Y, Xb);
  }

  // -------- decoder + pooling
  fill_f32<<<1, 64, 0, stream>>>(out, 0.0f, N_GRAPHS);
  decode_pool<<<nb(N_NODES, B), B, 0, stream>>>(Xb, dec_w, dec_b, batch, out);
}